// EdgeGAT_15616501088828
// MI455X (gfx1250) — compile-verified
//
#include <hip/hip_runtime.h>
#include <math.h>

#define NN 100000        // nodes
#define NE 1600000       // edges
#define FIN1 128         // NODE_F
#define FE 32            // EDGE_F
#define D 64             // HID == OUT
#define NEG_SLOPE 0.2f

typedef float v2f __attribute__((ext_vector_type(2)));
typedef float v8f __attribute__((ext_vector_type(8)));

// ---------------------------------------------------------------- WMMA helper
__device__ __forceinline__ v8f wmma4(v2f a, v2f b, v8f c) {
    // V_WMMA_F32_16X16X4_F32 : D = A(16x4) * B(4x16) + C(16x16), all fp32
    return __builtin_amdgcn_wmma_f32_16x16x4_f32(
        /*neg_a=*/false, a, /*neg_b=*/false, b,
        /*c_mod=*/(short)0, c, /*reuse_a=*/false, /*reuse_b=*/false);
}

// ordered-uint mapping so float max == uint max (monotone, handles negatives)
__device__ __forceinline__ unsigned f2o(float f) {
    unsigned b = __float_as_uint(f);
    return (b & 0x80000000u) ? ~b : (b | 0x80000000u);
}
__device__ __forceinline__ float o2f(unsigned u) {
    unsigned b = (u & 0x80000000u) ? (u & 0x7FFFFFFFu) : ~u;
    return __uint_as_float(b);
}

// ---------------------------------------------------------------- tiny: qe = We @ ae
__global__ void k_qe(const float* __restrict__ We, const float* __restrict__ ae,
                     float* __restrict__ qe) {
    int k = threadIdx.x;
    if (k < FE) {
        float s = 0.f;
        #pragma unroll 8
        for (int d = 0; d < D; ++d) s += We[k * D + d] * ae[d];
        qe[k] = s;
    }
}

// ---------------------------------------------------------------- sl = fs.al, sr = fs.ar
__global__ void k_node_logit(const float* __restrict__ fs,
                             const float* __restrict__ al, const float* __restrict__ ar,
                             float* __restrict__ sl, float* __restrict__ sr) {
    unsigned n = blockIdx.x * blockDim.x + threadIdx.x;
    if (n >= NN) return;
    const float4* row = reinterpret_cast<const float4*>(fs) + n * (D / 4);
    const float4* A = reinterpret_cast<const float4*>(al);
    const float4* B = reinterpret_cast<const float4*>(ar);
    float a = 0.f, b = 0.f;
    #pragma unroll
    for (int i = 0; i < D / 4; ++i) {
        float4 v = row[i], x = A[i], y = B[i];
        a += v.x * x.x + v.y * x.y + v.z * x.z + v.w * x.w;
        b += v.x * y.x + v.y * y.y + v.z * y.z + v.w * y.w;
    }
    sl[n] = a; sr[n] = b;
}

// ---------------------------------------------------------------- per-layer init
__global__ void k_init(float* __restrict__ acc, unsigned* __restrict__ emax_u,
                       float* __restrict__ den) {
    unsigned i = blockIdx.x * blockDim.x + threadIdx.x;
    if (i < NN * D) acc[i] = 0.f;
    if (i < NN) { emax_u[i] = 0u; den[i] = 0.f; }  // 0 == ordered(-inf) lower bound
}

// ---------------------------------------------------------------- edge logits + segment max
__global__ void k_edge_logit(const float* __restrict__ ef, const float* __restrict__ qe,
                             const float* __restrict__ sl, const float* __restrict__ sr,
                             const int* __restrict__ src, const int* __restrict__ dst,
                             float* __restrict__ elog, unsigned* __restrict__ emax_u) {
    unsigned e = blockIdx.x * blockDim.x + threadIdx.x;
    if (e >= NE) return;
    const float4* row = reinterpret_cast<const float4*>(ef) + e * (FE / 4);
    const float4* Q = reinterpret_cast<const float4*>(qe);
    float t = 0.f;
    #pragma unroll
    for (int i = 0; i < FE / 4; ++i) {
        float4 v = row[i], q = Q[i];
        t += v.x * q.x + v.y * q.y + v.z * q.z + v.w * q.w;
    }
    t += sl[(unsigned)src[e]] + sr[(unsigned)dst[e]];
    float l = t > 0.f ? t : NEG_SLOPE * t;          // leaky_relu
    elog[e] = l;
    atomicMax(&emax_u[(unsigned)dst[e]], f2o(l));
}

// ---------------------------------------------------------------- exp + segment sum
__global__ void k_edge_w(const float* __restrict__ elog, const int* __restrict__ dst,
                         const unsigned* __restrict__ emax_u,
                         float* __restrict__ wbuf, float* __restrict__ den) {
    unsigned e = blockIdx.x * blockDim.x + threadIdx.x;
    if (e >= NE) return;
    unsigned dv = (unsigned)dst[e];
    float w = __expf(elog[e] - o2f(emax_u[dv]));
    wbuf[e] = w;
    atomicAdd(&den[dv], w);
}

// ---------------------------------------------------------------- den -> 1/den (in place)
__global__ void k_rden(float* __restrict__ den) {
    unsigned n = blockIdx.x * blockDim.x + threadIdx.x;
    if (n >= NN) return;
    den[n] = 1.0f / den[n];     // one real divide per node, off the hot path
}

// ---------------------------------------------------------------- fused edge-GEMM + scatter
// per 16-edge tile: fe = ef_tile @ We via WMMA, then acc[dst] += alpha*(fs[src]+fe)
__global__ void __launch_bounds__(256) k_message(
    const float* __restrict__ ef, const float* __restrict__ We,
    const float* __restrict__ fs, const int* __restrict__ src,
    const int* __restrict__ dst, const float* __restrict__ wbuf,
    const float* __restrict__ rden, float* __restrict__ acc) {
    const int lane = threadIdx.x & 31;
    const int half = lane >> 4;       // which 16-lane half
    const int lr   = lane & 15;
    const int waveId = blockIdx.x * (blockDim.x >> 5) + (threadIdx.x >> 5);
    const int nwaves = gridDim.x * (blockDim.x >> 5);

    // Preload all B fragments of We[32x64] once per wave (8 k-steps x 4 n-tiles).
    // B[k][n]: lane = (n%16) + 16*(k>>1), vgpr = k&1 (mirror of A layout)
    v2f bfr[8][4];
    #pragma unroll
    for (int kk = 0; kk < 8; ++kk) {
        int k0 = kk * 4 + 2 * half;
        #pragma unroll
        for (int nt = 0; nt < 4; ++nt) {
            int n = nt * 16 + lr;
            v2f b; b.x = We[(unsigned)(k0 * D + n)]; b.y = We[(unsigned)((k0 + 1) * D + n)];
            bfr[kk][nt] = b;
        }
    }

    const int ntiles = NE / 16;       // 100000 exactly
    for (int tile = waveId; tile < ntiles; tile += nwaves) {
        const int rowbase = tile * 16;
        const unsigned arow = (unsigned)(rowbase + lr) * FE;   // 32-bit element offset

        // prefetch next grid-stride tile's A rows into cache (global_prefetch_b8)
        if (tile + nwaves < ntiles)
            __builtin_prefetch(ef + (unsigned)(rowbase + nwaves * 16 + lr) * FE, 0, 1);

        // per-tile row scalars: lanes 0..15 own one row each, broadcast later
        unsigned row16 = (unsigned)(rowbase + lr);
        int      s16   = src[row16];
        int      d16   = dst[row16];
        float    a16   = wbuf[row16] * rden[(unsigned)d16];    // alpha for this row

        v8f c0 = {0}, c1 = {0}, c2 = {0}, c3 = {0};
        #pragma unroll
        for (int kk = 0; kk < 8; ++kk) {
            unsigned k0 = (unsigned)(kk * 4 + 2 * half);       // A: M=lr, K=k0,k0+1
            v2f a = *(const v2f*)(ef + arow + k0);             // 8B contiguous load
            c0 = wmma4(a, bfr[kk][0], c0);
            c1 = wmma4(a, bfr[kk][1], c1);
            c2 = wmma4(a, bfr[kk][2], c2);
            c3 = wmma4(a, bfr[kk][3], c3);
        }
        // C layout: vgpr j holds row m = 8*half + j, column n = nt*16 + lr
        #pragma unroll
        for (int j = 0; j < 8; ++j) {
            int m = half * 8 + j;                    // source lane (0..15) for row m
            // 32-bit element offsets -> GVS addressing (sgpr base + vgpr offset)
            unsigned fso  = (unsigned)__shfl(s16, m) * D + (unsigned)lr;
            unsigned acco = (unsigned)__shfl(d16, m) * D + (unsigned)lr;
            float alpha = __shfl(a16, m);
            atomicAdd((float*)&acc[acco +  0], alpha * (c0[j] + fs[fso +  0]));
            atomicAdd((float*)&acc[acco + 16], alpha * (c1[j] + fs[fso + 16]));
            atomicAdd((float*)&acc[acco + 32], alpha * (c2[j] + fs[fso + 32]));
            atomicAdd((float*)&acc[acco + 48], alpha * (c3[j] + fs[fso + 48]));
        }
    }
}

// ---------------------------------------------------------------- node GEMM: out = h @ W
template <int FIN>
__global__ void __launch_bounds__(256) k_node_gemm(
    const float* __restrict__ h, const float* __restrict__ W, float* __restrict__ out) {
    __shared__ float WT[D * FIN];     // WT[n*FIN + k] = W[k*D + n] (transposed stage)
    for (int i = threadIdx.x; i < D * FIN; i += blockDim.x) {
        int n = i / FIN, k = i % FIN;
        WT[i] = W[(unsigned)(k * D + n)];
    }
    __syncthreads();
    const int lane = threadIdx.x & 31;
    const int half = lane >> 4;
    const int lr   = lane & 15;
    const int waveId = blockIdx.x * (blockDim.x >> 5) + (threadIdx.x >> 5);
    const int nwaves = gridDim.x * (blockDim.x >> 5);
    const int ntiles = NN / 16;       // 6250 exactly
    for (int tile = waveId; tile < ntiles; tile += nwaves) {
        const int rowbase = tile * 16;
        const unsigned arow = (unsigned)(rowbase + lr) * FIN;  // 32-bit element offset
        if (tile + nwaves < ntiles)
            __builtin_prefetch(h + (unsigned)(rowbase + nwaves * 16 + lr) * FIN, 0, 1);
        v8f c0 = {0}, c1 = {0}, c2 = {0}, c3 = {0};
        #pragma unroll 4
        for (int kk = 0; kk < FIN / 4; ++kk) {
            int k0 = kk * 4 + 2 * half;
            v2f a  = *(const v2f*)(h + arow + (unsigned)k0);
            v2f b0 = *(const v2f*)(&WT[(0 * 16 + lr) * FIN + k0]);
            v2f b1 = *(const v2f*)(&WT[(1 * 16 + lr) * FIN + k0]);
            v2f b2 = *(const v2f*)(&WT[(2 * 16 + lr) * FIN + k0]);
            v2f b3 = *(const v2f*)(&WT[(3 * 16 + lr) * FIN + k0]);
            c0 = wmma4(a, b0, c0);
            c1 = wmma4(a, b1, c1);
            c2 = wmma4(a, b2, c2);
            c3 = wmma4(a, b3, c3);
        }
        #pragma unroll
        for (int j = 0; j < 8; ++j) {
            unsigned o = (unsigned)(rowbase + half * 8 + j) * D + (unsigned)lr;
            out[o +  0] = c0[j];
            out[o + 16] = c1[j];
            out[o + 32] = c2[j];
            out[o + 48] = c3[j];
        }
    }
}

// ---------------------------------------------------------------- out = act(acc + res + b)
__global__ void k_finalize(const float* __restrict__ acc, const float* __restrict__ res,
                           const float* __restrict__ b, float* __restrict__ out,
                           int apply_tanh) {
    unsigned i = blockIdx.x * blockDim.x + threadIdx.x;
    if (i >= NN * D) return;
    float v = acc[i] + res[i] + b[i & (D - 1)];
    out[i] = apply_tanh ? tanhf(v) : v;
}

// ================================================================ host side
extern "C" void kernel_launch(void* const* d_in, const int* in_sizes, int n_in,
                              void* d_out, int out_size, void* d_ws, size_t ws_size,
                              hipStream_t stream) {
    // inputs in setup_inputs() order
    const float* node_in = (const float*)d_in[0];
    const float* edge_in = (const float*)d_in[1];
    const int*   src     = (const int*)d_in[2];
    const int*   dst     = (const int*)d_in[3];
    const float *W1 = (const float*)d_in[4],  *We1 = (const float*)d_in[5];
    const float *al1 = (const float*)d_in[6], *ar1 = (const float*)d_in[7];
    const float *ae1 = (const float*)d_in[8], *b1 = (const float*)d_in[9];
    const float *Wres1 = (const float*)d_in[10];
    const float *W2 = (const float*)d_in[11], *We2 = (const float*)d_in[12];
    const float *al2 = (const float*)d_in[13], *ar2 = (const float*)d_in[14];
    const float *ae2 = (const float*)d_in[15], *b2 = (const float*)d_in[16];
    const float *W3 = (const float*)d_in[17], *We3 = (const float*)d_in[18];
    const float *al3 = (const float*)d_in[19], *ar3 = (const float*)d_in[20];
    const float *ae3 = (const float*)d_in[21], *b3 = (const float*)d_in[22];

    // workspace carve (~142 MB)
    char* p = (char*)d_ws;
    auto take = [&](size_t bytes) {
        void* r = (void*)p;
        p += (bytes + 255) & ~(size_t)255;
        return r;
    };
    float*    fs     = (float*)take((size_t)NN * D * 4);
    float*    resbuf = (float*)take((size_t)NN * D * 4);
    float*    accum  = (float*)take((size_t)NN * D * 4);
    float*    hA     = (float*)take((size_t)NN * D * 4);
    float*    hB     = (float*)take((size_t)NN * D * 4);
    float*    sl     = (float*)take((size_t)NN * 4);
    float*    sr     = (float*)take((size_t)NN * 4);
    unsigned* emax_u = (unsigned*)take((size_t)NN * 4);
    float*    den    = (float*)take((size_t)NN * 4);
    float*    elog   = (float*)take((size_t)NE * 4);
    float*    wbuf   = (float*)take((size_t)NE * 4);
    float*    qe     = (float*)take(FE * 4);

    const int gN  = (NN + 255) / 256;
    const int gND = ((size_t)NN * D + 255) / 256;
    const int gE  = (NE + 255) / 256;

    auto layer = [&](const float* h, int fin, const float* W, const float* We,
                     const float* al, const float* ar, const float* ae,
                     const float* bb, const float* Wres, float* hout, int act) {
        k_qe<<<1, FE, 0, stream>>>(We, ae, qe);
        if (fin == FIN1) k_node_gemm<FIN1><<<512, 256, 0, stream>>>(h, W, fs);
        else             k_node_gemm<D><<<512, 256, 0, stream>>>(h, W, fs);
        const float* resptr = h;
        if (Wres) { k_node_gemm<FIN1><<<512, 256, 0, stream>>>(h, Wres, resbuf); resptr = resbuf; }
        k_node_logit<<<gN, 256, 0, stream>>>(fs, al, ar, sl, sr);
        k_init<<<gND, 256, 0, stream>>>(accum, emax_u, den);
        k_edge_logit<<<gE, 256, 0, stream>>>(edge_in, qe, sl, sr, src, dst, elog, emax_u);
        k_edge_w<<<gE, 256, 0, stream>>>(elog, dst, emax_u, wbuf, den);
        k_rden<<<gN, 256, 0, stream>>>(den);
        k_message<<<1250, 256, 0, stream>>>(edge_in, We, fs, src, dst, wbuf, den, accum);
        k_finalize<<<gND, 256, 0, stream>>>(accum, resptr, bb, hout, act);
    };

    layer(node_in, FIN1, W1, We1, al1, ar1, ae1, b1, Wres1, hA, 1);  // layer 1 + tanh
    layer(hA, D, W2, We2, al2, ar2, ae2, b2, nullptr, hB, 1);        // hidden 1 + tanh
    layer(hB, D, W2, We2, al2, ar2, ae2, b2, nullptr, hA, 1);        // hidden 2 + tanh
    layer(hA, D, W3, We3, al3, ar3, ae3, b3, nullptr, (float*)d_out, 0); // output layer
}